// QuantumHybridQLSTM_65481071410675
// MI455X (gfx1250) — compile-verified
//
#include <hip/hip_runtime.h>

// QLSTM with 4-qubit quantum layers analytically reduced to cosine products.
// WMMA with A = weights (16 rows = 4 gates x 4 wires), B = per-batch comb vectors
// (16 batch columns). D gives each lane one batch column: lanes 0-15 rows 0-7 (f,i),
// lanes 16-31 rows 8-15 (u,o) of batch lane&15. All gate-internal math is in-register;
// only 10 half-swap shuffles per step. No LDS, no divergent selects.

typedef __attribute__((ext_vector_type(2))) float v2f;
typedef __attribute__((ext_vector_type(8))) float v8f;

#define S_LEN 512
#define B_LEN 16384

__device__ __forceinline__ float sigm(float xv) {
    // 1 / (1 + e^-x) via v_exp_f32 + v_rcp_f32
    return __builtin_amdgcn_rcpf(1.0f + __expf(-xv));
}

__global__ __launch_bounds__(256) void qlstm_wmma_kernel(
    const float* __restrict__ x,
    const float* __restrict__ Wf, const float* __restrict__ bf,
    const float* __restrict__ Wi, const float* __restrict__ bi,
    const float* __restrict__ Wu, const float* __restrict__ bu,
    const float* __restrict__ Wo, const float* __restrict__ bo,
    const float* __restrict__ pf, const float* __restrict__ pi_,
    const float* __restrict__ pu, const float* __restrict__ po,
    float* __restrict__ ys, float* __restrict__ hx_out, float* __restrict__ cx_out)
{
    const int tid    = threadIdx.x;
    const int lane   = tid & 31;
    const int waveId = blockIdx.x * 8 + (tid >> 5);
    const int batch0 = waveId * 16;
    const int m      = lane & 15;        // batch column owned by this lane (N index)
    const bool lo    = (lane < 16);

    // --- static A matrix = Wall (16 rows x K=5), ISA 16x4 fp32 A layout:
    //     lanes 0-15: M=lane, VGPR0=K0, VGPR1=K1 ; lanes 16-31: M=lane-16, K2, K3
    {
    }
    const int arow  = lane & 15;          // output row index n = 4*gate + wire
    const int agate = arow >> 2;
    const int awire = arow & 3;
    const float* Wg = (agate == 0) ? Wf : (agate == 1) ? Wi : (agate == 2) ? Wu : Wo;
    const float* rp = Wg + awire * 5;     // weights over comb = [conv, h0, h1, h2, h3]
    v2f A1, A2;
    A1.x = lo ? rp[0] : rp[2];            // K=0 / K=2
    A1.y = lo ? rp[1] : rp[3];            // K=1 / K=3
    A2.x = lo ? rp[4] : 0.0f;             // K=4 / K=6(pad)
    A2.y = 0.0f;                          // K=5,7 pad

    // --- bias (b + qp folded) in C accumulator: C[row=v+8*half][*] ---
    v8f biasC;
#pragma unroll
    for (int v = 0; v < 4; ++v) {
        biasC[v]     = lo ? (bf[v] + pf[v])  : (bu[v] + pu[v]);   // rows v (f) / v+8 (u)
        biasC[v + 4] = lo ? (bi[v] + pi_[v]) : (bo[v] + po[v]);   // rows v+4 (i) / v+12 (o)
    }

    // activation coefficients for row group v0-3: sigmoid (lo: f) vs tanh (hi: u)
    //   out = sigm(e*sc_in)*sc_out + sc_off     (tanh(x) = 2*sigm(2x)-1)
    const float sc_in  = lo ? 1.0f : 2.0f;
    const float sc_out = lo ? 1.0f : 2.0f;
    const float sc_off = lo ? 0.0f : -1.0f;

    // --- LSTM state (valid at lo lanes; hi lanes compute don't-care copies) ---
    float cst[4] = {0.f, 0.f, 0.f, 0.f};
    float h[4]   = {0.f, 0.f, 0.f, 0.f};
    float h1s = 0.f, h2s = 0.f;           // lo-half h1,h2 shuffled to hi lanes

    const float4* xq = (const float4*)x;
    float4 xv = make_float4(0.f, 0.f, 0.f, 0.f);
    if (lo) xv = xq[batch0 + m];          // pipelined x load for t=0

    for (int t = 0; t < S_LEN; ++t) {
        const float4 xc = xv;
        if (lo) {
            if (t + 1 < S_LEN) xv = xq[(size_t)(t + 1) * B_LEN + batch0 + m];
            if (t + 4 < S_LEN) __builtin_prefetch(&xq[(size_t)(t + 4) * B_LEN + batch0 + m], 0, 1);
        }
        const float conv = 0.25f * ((xc.x > 0.5f ? 1.f : 0.f) + (xc.y > 0.5f ? 1.f : 0.f) +
                                    (xc.z > 0.5f ? 1.f : 0.f) + (xc.w > 0.5f ? 1.f : 0.f));

        // B matrix (K x 16 batch): lanes 0-15 hold K=0,1 ; lanes 16-31 hold K=2,3
        v2f B1, B2;
        B1.x = lo ? conv : h1s;           // comb[0]=conv  | comb[2]=h1
        B1.y = lo ? h[0] : h2s;           // comb[1]=h0    | comb[3]=h2
        B2.x = lo ? h[3] : 0.0f;          // comb[4]=h3    | pad
        B2.y = 0.0f;

        // angles(16x16) = Wall(16x5) @ comb^T(5x16) + (b+qp); K=5 as two K=4 WMMAs
        v8f d = __builtin_amdgcn_wmma_f32_16x16x4_f32(false, A1, false, B1, (short)0, biasC, false, false);
        d     = __builtin_amdgcn_wmma_f32_16x16x4_f32(false, A2, false, B2, (short)0, d,     false, false);

        // cosines of all 8 rows this lane holds
        float c[8];
#pragma unroll
        for (int v = 0; v < 8; ++v) c[v] = __cosf(d[v]);

        // quantum layer: per gate group of 4 rows, in-register cosine products
        //   e0 = c1 c2 c3, e1 = c0 c1, e2 = c0 c1 c2, e3 = c0 c1 c2 c3
        float e[8];
#pragma unroll
        for (int g2 = 0; g2 < 2; ++g2) {
            const int b = g2 * 4;
            const float p01 = c[b] * c[b + 1];
            const float c23 = c[b + 2] * c[b + 3];
            e[b + 0] = c[b + 1] * c23;
            e[b + 1] = p01;
            e[b + 2] = p01 * c[b + 2];
            e[b + 3] = p01 * c23;
        }

        // activations: group v0-3 = f(lo)/u-tanh(hi) via coefficients; v4-7 = sigmoid (i/o)
        float a[8];
#pragma unroll
        for (int v = 0; v < 4; ++v) {
            a[v]     = sigm(e[v] * sc_in) * sc_out + sc_off;
            a[v + 4] = sigm(e[v + 4]);
        }

        // half-swap: lo lanes fetch hi's g (tanh u) and o (sigmoid) values
        float gg[4], oo[4];
#pragma unroll
        for (int v = 0; v < 4; ++v) {
            gg[v] = __shfl_xor(a[v], 16, 32);
            oo[v] = __shfl_xor(a[v + 4], 16, 32);
        }

        // LSTM cell update (valid on lo lanes)
#pragma unroll
        for (int v = 0; v < 4; ++v) {
            const float cn = a[v] * cst[v] + a[v + 4] * gg[v];   // f*c + i*g
            cst[v] = cn;
            float th = sigm(cn + cn); th = th + th - 1.0f;        // tanh(cn)
            h[v] = oo[v] * th;
        }

        // feed h1,h2 to hi lanes for next step's B matrix
        h1s = __shfl_xor(h[1], 16, 32);
        h2s = __shfl_xor(h[2], 16, 32);

        if (lo) {
            ((float4*)ys)[(size_t)t * B_LEN + batch0 + m] = make_float4(h[0], h[1], h[2], h[3]);
        }
    }

    if (lo) {
        ((float4*)hx_out)[batch0 + m] = make_float4(h[0], h[1], h[2], h[3]);
        ((float4*)cx_out)[batch0 + m] = make_float4(cst[0], cst[1], cst[2], cst[3]);
    }
}

extern "C" void kernel_launch(void* const* d_in, const int* in_sizes, int n_in,
                              void* d_out, int out_size, void* d_ws, size_t ws_size,
                              hipStream_t stream) {
    (void)in_sizes; (void)n_in; (void)out_size; (void)d_ws; (void)ws_size;
    const float* x   = (const float*)d_in[0];
    const float* Wf  = (const float*)d_in[1];
    const float* bf  = (const float*)d_in[2];
    const float* Wi  = (const float*)d_in[3];
    const float* bi  = (const float*)d_in[4];
    const float* Wu  = (const float*)d_in[5];
    const float* bu  = (const float*)d_in[6];
    const float* Wo  = (const float*)d_in[7];
    const float* bo  = (const float*)d_in[8];
    const float* pf  = (const float*)d_in[9];
    const float* pi_ = (const float*)d_in[10];
    const float* pu  = (const float*)d_in[11];
    const float* po  = (const float*)d_in[12];

    float* out = (float*)d_out;
    float* ys  = out;
    float* hx  = ys + (size_t)S_LEN * B_LEN * 4;
    float* cx  = hx + (size_t)B_LEN * 4;

    dim3 grid(B_LEN / 128);   // 128 blocks * 8 waves * 16 batch = 16384
    dim3 block(256);
    qlstm_wmma_kernel<<<grid, block, 0, stream>>>(x, Wf, bf, Wi, bi, Wu, bu, Wo, bo,
                                                  pf, pi_, pu, po, ys, hx, cx);
}